// SimpleGNN_51204600103280
// MI455X (gfx1250) — compile-verified
//
#include <hip/hip_runtime.h>
#include <hip/hip_bf16.h>

#define NN 100000
#define NE 1200000
#define NG 512
#define IN_DIM 16
#define HID 64
#define EPS 1e-5f

typedef float v2f __attribute__((ext_vector_type(2)));
typedef float v8f __attribute__((ext_vector_type(8)));

static __device__ __forceinline__ int gtid() { return blockIdx.x * blockDim.x + threadIdx.x; }

// ---------------- utility ----------------
__global__ void k_zero(float* __restrict__ p, int n) {
    int i = gtid(); if (i < n) p[i] = 0.0f;
}

// ---------------- degree / edge coefficients ----------------
__global__ void k_deg_init(float* __restrict__ deg, int n) {
    int i = gtid(); if (i < n) deg[i] = 1.0f;   // self-loop contributes 1
}
__global__ void k_deg_count(const int* __restrict__ dst, float* __restrict__ deg, int e) {
    int i = gtid(); if (i < e) atomicAdd(&deg[dst[i]], 1.0f);
}
__global__ void k_dis(float* __restrict__ deg_dis, int n) {
    int i = gtid(); if (i < n) { float d = deg_dis[i]; deg_dis[i] = d > 0.0f ? rsqrtf(d) : 0.0f; }
}
__global__ void k_coef(const int* __restrict__ src, const int* __restrict__ dst,
                       const float* __restrict__ dis, float* __restrict__ coef, int e) {
    int i = gtid(); if (i < e) coef[i] = dis[src[i]] * dis[dst[i]];
}

// ---------------- batchnorm (training-mode, biased var) ----------------
__global__ void k_bn_stats(const float* __restrict__ X, int n, int C,
                           float* __restrict__ sum, float* __restrict__ sumsq) {
    __shared__ float ls[256], lq[256];
    int c  = threadIdx.x & (C - 1);
    int rg = gtid() / C;
    int nrg = (gridDim.x * blockDim.x) / C;
    float s = 0.0f, q = 0.0f;
    for (int i = rg; i < n; i += nrg) {
        float v = X[(size_t)i * C + c];
        s += v; q += v * v;
    }
    ls[threadIdx.x] = s; lq[threadIdx.x] = q;
    __syncthreads();
    if (threadIdx.x < (unsigned)C) {
        float S = 0.0f, Q = 0.0f;
        for (int t = threadIdx.x; t < 256; t += C) { S += ls[t]; Q += lq[t]; }
        atomicAdd(&sum[c], S);
        atomicAdd(&sumsq[c], Q);
    }
}

__global__ void k_bn_apply(const float* __restrict__ X, float* __restrict__ Y, int n, int C,
                           const float* __restrict__ sum, const float* __restrict__ sumsq,
                           const float* __restrict__ g, const float* __restrict__ b, int do_relu) {
    int idx = gtid();
    if (idx >= n * C) return;
    int c = idx & (C - 1);
    float invn = 1.0f / (float)n;
    float m = sum[c] * invn;
    float v = sumsq[c] * invn - m * m;
    float r = rsqrtf(v + EPS);
    float y = (X[idx] - m) * r * g[c] + b[c];
    if (do_relu) y = fmaxf(y, 0.0f);
    Y[idx] = y;
}

// ---------------- dense GEMM via V_WMMA_F32_16X16X4_F32 ----------------
// A [nrows, CIN] row-major f32, W [CIN, cout] row-major f32 -> T [nrows, cout].
// Each wave owns one 16-col n-tile, preloads B fragments for the whole K dim
// into VGPRs (CIN/4 x v2f), then grid-strides over 16-row chunks:
// inner loop = one b64 A-load + one WMMA per k-step.
template <int CIN>
__global__ void k_gemm_wmma(const float* __restrict__ A, const float* __restrict__ W,
                            float* __restrict__ T, int nrows, int cout) {
    constexpr int KS = CIN / 4;                  // k-steps
    int w    = gtid() >> 5;
    int lane = threadIdx.x & 31;
    int nt   = cout >> 4;                        // number of 16-col tiles
    int ntile  = w % nt;
    int chunk0 = w / nt;
    int nchunks = nrows >> 4;
    int total_waves = (gridDim.x * blockDim.x) >> 5;
    int cstride = total_waves / nt;
    int half = lane >> 4;                        // 0: lanes 0-15, 1: lanes 16-31
    int l16  = lane & 15;
    int n0   = ntile * 16;

    // Preload B fragments (4x16 f32 per k-step): VGPR0 = K=(4k+2*half), VGPR1 = next K.
    v2f breg[KS];
#pragma unroll
    for (int kk = 0; kk < KS; ++kk) {
        int ka = kk * 4 + 2 * half;
        breg[kk][0] = W[(size_t)ka * cout + n0 + l16];
        breg[kk][1] = W[(size_t)(ka + 1) * cout + n0 + l16];
    }

    for (int c = chunk0; c < nchunks; c += cstride) {
        int row0 = c << 4;
        // A 16x4 f32 layout: lane holds row M=l16, K pair starting at 2*half.
        const float* arow = A + (size_t)(row0 + l16) * CIN + 2 * half;
        v8f acc = {};
#pragma unroll
        for (int kk = 0; kk < KS; ++kk) {
            v2f a = *(const v2f*)(arow + kk * 4);   // 8B-aligned contiguous K pair
            acc = __builtin_amdgcn_wmma_f32_16x16x4_f32(
                false, a, false, breg[kk], (short)0, acc, false, false);
        }
        // C/D 16x16 f32: VGPR r holds M = r + 8*half, N = l16
        float* trow = T + (size_t)row0 * cout + n0 + l16;
#pragma unroll
        for (int r = 0; r < 8; ++r)
            trow[(size_t)(r + 8 * half) * cout] = acc[r];
    }
}

// ---------------- GCN aggregation ----------------
// agg[i][c] = dis[i]^2 * t[i][c] + bias[c]   (self loop + bias)
__global__ void k_agg_init(const float* __restrict__ t, const float* __restrict__ dis,
                           const float* __restrict__ bias, float* __restrict__ agg, int n) {
    int idx = gtid();
    if (idx >= n * HID) return;
    int i = idx >> 6, c = idx & 63;
    float d = dis[i];
    agg[idx] = d * d * t[idx] + bias[c];
}

// one wave per edge, 2 features per lane: agg[dst] += coef * t[src]
__global__ void k_agg_edges(const int* __restrict__ src, const int* __restrict__ dst,
                            const float* __restrict__ coef, const float* __restrict__ t,
                            float* __restrict__ agg, int e) {
    int tid = gtid();
    int ei = tid >> 5, lane = tid & 31;
    if (ei >= e) return;
    int s = src[ei], d = dst[ei];
    float w = coef[ei];
    const float* ts = t + (size_t)s * HID;
    float* ad = agg + (size_t)d * HID;
    atomicAdd(&ad[lane],      w * ts[lane]);
    atomicAdd(&ad[lane + 32], w * ts[lane + 32]);
}

// ---------------- global mean pool + head ----------------
__global__ void k_pool(const float* __restrict__ h, const int* __restrict__ bid,
                       float* __restrict__ pooled, float* __restrict__ cnt, int n) {
    int idx = gtid();
    if (idx >= n * HID) return;
    int i = idx >> 6, c = idx & 63;
    int g = bid[i];
    atomicAdd(&pooled[(size_t)g * HID + c], h[idx]);
    if (c == 0) atomicAdd(&cnt[g], 1.0f);
}

__global__ void k_head(const float* __restrict__ pooled, const float* __restrict__ cnt,
                       const float* __restrict__ Wc1, const float* __restrict__ bc1,
                       const float* __restrict__ Wc2, const float* __restrict__ bc2,
                       float* __restrict__ out) {
    int g = gtid();
    if (g >= NG) return;
    float inv = 1.0f / fmaxf(cnt[g], 1.0f);
    float p[HID];
#pragma unroll
    for (int k = 0; k < HID; ++k) p[k] = pooled[(size_t)g * HID + k] * inv;
    float o0 = bc2[0], o1 = bc2[1];
    for (int j = 0; j < HID; ++j) {
        float z = bc1[j];
#pragma unroll
        for (int k = 0; k < HID; ++k) z += p[k] * Wc1[k * HID + j];
        z = fmaxf(z, 0.0f);
        o0 += z * Wc2[j * 2 + 0];
        o1 += z * Wc2[j * 2 + 1];
    }
    out[g * 2 + 0] = o0;
    out[g * 2 + 1] = o1;
}

// ---------------- orchestration ----------------
static inline int cdiv(int a, int b) { return (a + b - 1) / b; }

extern "C" void kernel_launch(void* const* d_in, const int* in_sizes, int n_in,
                              void* d_out, int out_size, void* d_ws, size_t ws_size,
                              hipStream_t stream) {
    const float* x       = (const float*)d_in[0];
    const int*   ei      = (const int*)d_in[1];
    const int*   bid     = (const int*)d_in[2];
    const float* bn_in_g = (const float*)d_in[3];
    const float* bn_in_b = (const float*)d_in[4];
    const float* Ws[3]  = { (const float*)d_in[5],  (const float*)d_in[9],  (const float*)d_in[13] };
    const float* bs[3]  = { (const float*)d_in[6],  (const float*)d_in[10], (const float*)d_in[14] };
    const float* gs[3]  = { (const float*)d_in[7],  (const float*)d_in[11], (const float*)d_in[15] };
    const float* bes[3] = { (const float*)d_in[8],  (const float*)d_in[12], (const float*)d_in[16] };
    const float* Wc1 = (const float*)d_in[17];
    const float* bc1 = (const float*)d_in[18];
    const float* Wc2 = (const float*)d_in[19];
    const float* bc2 = (const float*)d_in[20];

    const int N = in_sizes[0] / IN_DIM;
    const int E = in_sizes[1] / 2;
    const int* src = ei;
    const int* dst = ei + E;

    float* ws     = (float*)d_ws;
    float* bufA   = ws;                                  // N*HID
    float* bufB   = bufA + (size_t)N * HID;              // N*HID
    float* dis    = bufB + (size_t)N * HID;              // N  (deg -> dis in place)
    float* coef   = dis + N;                             // E
    float* stats  = coef + E;                            // 2*HID (sum, sumsq)
    float* pooled = stats + 2 * HID;                     // NG*HID
    float* cnt    = pooled + (size_t)NG * HID;           // NG

    // degree & symmetric-norm edge coefficients (shared by all 3 layers)
    k_deg_init <<<cdiv(N, 256), 256, 0, stream>>>(dis, N);
    k_deg_count<<<cdiv(E, 256), 256, 0, stream>>>(dst, dis, E);
    k_dis      <<<cdiv(N, 256), 256, 0, stream>>>(dis, N);
    k_coef     <<<cdiv(E, 256), 256, 0, stream>>>(src, dst, dis, coef, E);

    // input batchnorm: x [N,16] -> bufA [N,16]
    k_zero    <<<1, 256, 0, stream>>>(stats, 2 * HID);
    k_bn_stats<<<512, 256, 0, stream>>>(x, N, IN_DIM, stats, stats + HID);
    k_bn_apply<<<cdiv(N * IN_DIM, 256), 256, 0, stream>>>(x, bufA, N, IN_DIM,
                                                          stats, stats + HID, bn_in_g, bn_in_b, 0);

    const int GEMM_BLOCKS = 512;  // 4096 waves, divisible by 4 n-tiles
    for (int l = 0; l < 3; ++l) {
        // GEMM: bufA [N,cin] @ W [cin,64] -> bufB [N,64]  (WMMA f32 16x16x4)
        if (l == 0)
            k_gemm_wmma<IN_DIM><<<GEMM_BLOCKS, 256, 0, stream>>>(bufA, Ws[l], bufB, N, HID);
        else
            k_gemm_wmma<HID>  <<<GEMM_BLOCKS, 256, 0, stream>>>(bufA, Ws[l], bufB, N, HID);
        // aggregate (self-loop + bias init, then edge atomics): bufB -> bufA
        k_agg_init <<<cdiv(N * HID, 256), 256, 0, stream>>>(bufB, dis, bs[l], bufA, N);
        k_agg_edges<<<cdiv(E * 32, 256), 256, 0, stream>>>(src, dst, coef, bufB, bufA, E);
        // batchnorm + relu in place on bufA
        k_zero    <<<1, 256, 0, stream>>>(stats, 2 * HID);
        k_bn_stats<<<512, 256, 0, stream>>>(bufA, N, HID, stats, stats + HID);
        k_bn_apply<<<cdiv(N * HID, 256), 256, 0, stream>>>(bufA, bufA, N, HID,
                                                           stats, stats + HID, gs[l], bes[l], 1);
    }

    // global mean pool + classifier head
    k_zero<<<cdiv(NG * HID + NG, 256), 256, 0, stream>>>(pooled, NG * HID + NG);
    k_pool<<<cdiv(N * HID, 256), 256, 0, stream>>>(bufA, bid, pooled, cnt, N);
    k_head<<<cdiv(NG, 256), 256, 0, stream>>>(pooled, cnt, Wc1, bc1, Wc2, bc2, (float*)d_out);
}